// Model_35828617183373
// MI455X (gfx1250) — compile-verified
//
#include <hip/hip_runtime.h>
#include <math.h>

// Problem constants (match reference)
constexpr int IMG = 256;
constexpr int M   = 1920;
constexpr int B   = 64;
constexpr int MT  = M / 16;          // 120 m-tiles
constexpr float TWO_PI = 6.283185307179586f;

typedef __attribute__((ext_vector_type(2))) float v2f;
typedef __attribute__((ext_vector_type(8))) float v8f;

// D = A(16x4 f32) * B(4x16 f32) + C(16x16 f32)
__device__ __forceinline__ v8f wmma4(v2f a, v2f b, v8f c) {
    return __builtin_amdgcn_wmma_f32_16x16x4_f32(
        /*neg_a=*/false, a, /*neg_b=*/false, b,
        /*c_mod=*/(short)0, c, /*reuse_a=*/false, /*reuse_b=*/false);
}

// A fragment: rows m0..m0+15 (lane%16), K slice k0..k0+3.
// lanes 0-15: VGPR0=K0, VGPR1=K1 ; lanes 16-31: VGPR0=K2, VGPR1=K3
__device__ __forceinline__ v2f ldA(const float* Mat, int ld, int row0, int k0,
                                   int r, int half) {
    const float* p = Mat + (size_t)(row0 + r) * ld + k0 + 2 * half;
    v2f a = {p[0], p[1]};
    return a;
}

// B fragment: K slice k0..k0+3 (rows), cols n0..n0+15 (lane%16).
// lanes 0-15: VGPR0=K0 row, VGPR1=K1 row ; lanes 16-31: K2, K3
__device__ __forceinline__ v2f ldB(const float* Mat, int ld, int k0, int col0,
                                   int r, int half) {
    const float* p = Mat + (size_t)(k0 + 2 * half) * ld + col0 + r;
    v2f b = {p[0], p[ld]};
    return b;
}

// ---------------------------------------------------------------------------
// Kernel 1: phase matrices  Ax = exp(-2i*pi*kx*g),  Ay = exp(-2i*pi*ky*g)
// ---------------------------------------------------------------------------
__global__ void phase_kernel(const float* __restrict__ samples,
                             float* __restrict__ AxR, float* __restrict__ AxI,
                             float* __restrict__ AyR, float* __restrict__ AyI) {
    int idx = blockIdx.x * blockDim.x + threadIdx.x;
    if (idx >= M * IMG) return;
    int m = idx / IMG;
    int i = idx - m * IMG;
    float g  = (float)i - (float)(IMG / 2);
    float kx = samples[2 * m + 0];
    float ky = samples[2 * m + 1];
    float s, c;
    sincosf(-TWO_PI * kx * g, &s, &c);
    AxR[idx] = c; AxI[idx] = s;
    sincosf(-TWO_PI * ky * g, &s, &c);
    AyR[idx] = c; AyI[idx] = s;
}

// ---------------------------------------------------------------------------
// Kernel 2: fused forward NDFT + Ay contraction + density weighting.
// One wave per (batch, 16-row m-tile), blocked 2 j-tiles deep so each Ax
// fragment feeds 8 WMMAs. Epilogue contracts the register tiles with Ay and
// reduces row partials across the 16 column lanes.
// ---------------------------------------------------------------------------
__global__ void __launch_bounds__(256)
fwd_kernel(const float* __restrict__ xR, const float* __restrict__ xI,
           const float* __restrict__ AxR, const float* __restrict__ AxI,
           const float* __restrict__ AyR, const float* __restrict__ AyI,
           const float* __restrict__ density,
           float* __restrict__ kwR, float* __restrict__ kwI) {
    int wave = (blockIdx.x * blockDim.x + threadIdx.x) >> 5;
    if (wave >= B * MT) return;                 // wave-uniform guard
    int b  = wave / MT;
    int m0 = (wave - b * MT) * 16;
    int lane = threadIdx.x & 31;
    int r = lane & 15, half = lane >> 4;

    const float* xRb = xR + (size_t)b * IMG * IMG;
    const float* xIb = xI + (size_t)b * IMG * IMG;

    float partR[8], partI[8];
#pragma unroll
    for (int v = 0; v < 8; ++v) { partR[v] = 0.f; partI[v] = 0.f; }

    for (int jt = 0; jt < IMG; jt += 32) {      // 2 j-tiles per iteration
        v8f RR0 = {}, II0 = {}, RI0 = {}, IR0 = {};
        v8f RR1 = {}, II1 = {}, RI1 = {}, IR1 = {};
        for (int k0 = 0; k0 < IMG; k0 += 4) {
            v2f aR  = ldA(AxR, IMG, m0, k0, r, half);
            v2f aI  = ldA(AxI, IMG, m0, k0, r, half);
            v2f bR0 = ldB(xRb, IMG, k0, jt,      r, half);
            v2f bI0 = ldB(xIb, IMG, k0, jt,      r, half);
            v2f bR1 = ldB(xRb, IMG, k0, jt + 16, r, half);
            v2f bI1 = ldB(xIb, IMG, k0, jt + 16, r, half);
            RR0 = wmma4(aR, bR0, RR0);
            II0 = wmma4(aI, bI0, II0);
            RI0 = wmma4(aR, bI0, RI0);
            IR0 = wmma4(aI, bR0, IR0);
            RR1 = wmma4(aR, bR1, RR1);
            II1 = wmma4(aI, bI1, II1);
            RI1 = wmma4(aR, bI1, RI1);
            IR1 = wmma4(aI, bR1, IR1);
        }
        // t = Ax*x tiles; contract with Ay over both 16-column j-tiles.
        // C/D layout: VGPR v of lane -> row = v + 8*half, col = r.
#pragma unroll
        for (int v = 0; v < 8; ++v) {
            int row  = m0 + v + 8 * half;
            const float* ayRrow = AyR + (size_t)row * IMG;
            const float* ayIrow = AyI + (size_t)row * IMG;
            {
                int col  = jt + r;
                float tR = RR0[v] - II0[v];
                float tI = RI0[v] + IR0[v];
                float aR = ayRrow[col], aI = ayIrow[col];
                partR[v] += aR * tR - aI * tI;
                partI[v] += aR * tI + aI * tR;
            }
            {
                int col  = jt + 16 + r;
                float tR = RR1[v] - II1[v];
                float tI = RI1[v] + IR1[v];
                float aR = ayRrow[col], aI = ayIrow[col];
                partR[v] += aR * tR - aI * tI;
                partI[v] += aR * tI + aI * tR;
            }
        }
    }
    // Reduce across the 16 columns held by lanes within each half-wave.
#pragma unroll
    for (int v = 0; v < 8; ++v) {
        float sR = partR[v], sI = partI[v];
#pragma unroll
        for (int off = 8; off >= 1; off >>= 1) {
            sR += __shfl_xor(sR, off, 16);
            sI += __shfl_xor(sI, off, 16);
        }
        if (r == 0) {
            int m = m0 + v + 8 * half;
            float d = density[m];
            kwR[(size_t)b * M + m] = sR * d;
            kwI[(size_t)b * M + m] = sI * d;
        }
    }
}

// ---------------------------------------------------------------------------
// Kernel 3: adjoint NDFT.  adj[b] = (conj(Ax)*kw)^T (IMGxM) x conj(Ay) (MxIMG)
// One wave per (batch, 16x32 output strip): the kw-weighted conj(Ax) fragment
// (the expensive per-k VALU part) feeds 8 WMMAs. conj(Ay) is handled by sign
// flips in the epilogue instead of per-iteration negation:
//   real = RR + II,  imag = IR - RI   (with B fragments built from +AyI).
// ---------------------------------------------------------------------------
__global__ void __launch_bounds__(256)
adj_kernel(const float* __restrict__ AxR, const float* __restrict__ AxI,
           const float* __restrict__ AyR, const float* __restrict__ AyI,
           const float* __restrict__ kwR, const float* __restrict__ kwI,
           float* __restrict__ out) {
    int wave = (blockIdx.x * blockDim.x + threadIdx.x) >> 5;
    if (wave >= B * 16 * 8) return;             // wave-uniform guard
    int b  = wave >> 7;
    int t  = wave & 127;
    int i0 = (t >> 3) << 4;                     // 16 i-tiles
    int j0 = (t & 7) << 5;                      // 8 strips of 2 j-tiles
    int lane = threadIdx.x & 31;
    int r = lane & 15, half = lane >> 4;

    const float* kwRb = kwR + (size_t)b * M;
    const float* kwIb = kwI + (size_t)b * M;

    v8f RR0 = {}, II0 = {}, RI0 = {}, IR0 = {};
    v8f RR1 = {}, II1 = {}, RI1 = {}, IR1 = {};
    for (int m0 = 0; m0 < M; m0 += 4) {
        int mk0 = m0 + 2 * half;
        int mk1 = mk0 + 1;
        // A[i, m] = conj(Ax[m, i]) * kw[m], i = i0 + r (coalesced over lanes)
        float axR0 = AxR[(size_t)mk0 * IMG + i0 + r];
        float axI0 = AxI[(size_t)mk0 * IMG + i0 + r];
        float axR1 = AxR[(size_t)mk1 * IMG + i0 + r];
        float axI1 = AxI[(size_t)mk1 * IMG + i0 + r];
        float wR0 = kwRb[mk0], wI0 = kwIb[mk0];
        float wR1 = kwRb[mk1], wI1 = kwIb[mk1];
        v2f aR = { axR0 * wR0 + axI0 * wI0,  axR1 * wR1 + axI1 * wI1 };
        v2f aI = { axR0 * wI0 - axI0 * wR0,  axR1 * wI1 - axI1 * wR1 };
        // B fragments from +Ay (conj applied via epilogue signs)
        v2f bR0 = { AyR[(size_t)mk0 * IMG + j0 + r],
                    AyR[(size_t)mk1 * IMG + j0 + r] };
        v2f bI0 = { AyI[(size_t)mk0 * IMG + j0 + r],
                    AyI[(size_t)mk1 * IMG + j0 + r] };
        v2f bR1 = { AyR[(size_t)mk0 * IMG + j0 + 16 + r],
                    AyR[(size_t)mk1 * IMG + j0 + 16 + r] };
        v2f bI1 = { AyI[(size_t)mk0 * IMG + j0 + 16 + r],
                    AyI[(size_t)mk1 * IMG + j0 + 16 + r] };
        RR0 = wmma4(aR, bR0, RR0);
        II0 = wmma4(aI, bI0, II0);
        RI0 = wmma4(aR, bI0, RI0);
        IR0 = wmma4(aI, bR0, IR0);
        RR1 = wmma4(aR, bR1, RR1);
        II1 = wmma4(aI, bI1, II1);
        RI1 = wmma4(aR, bI1, RI1);
        IR1 = wmma4(aI, bR1, IR1);
    }
    const float scale = 1.0f / ((float)IMG * (float)IMG);
#pragma unroll
    for (int v = 0; v < 8; ++v) {
        int i = i0 + v + 8 * half;
        {
            int j = j0 + r;
            float oR = (RR0[v] + II0[v]) * scale;
            float oI = (IR0[v] - RI0[v]) * scale;
            size_t base = (((size_t)b * IMG + i) * IMG + j) * 2;
            out[base + 0] = oR;
            out[base + 1] = oI;
        }
        {
            int j = j0 + 16 + r;
            float oR = (RR1[v] + II1[v]) * scale;
            float oI = (IR1[v] - RI1[v]) * scale;
            size_t base = (((size_t)b * IMG + i) * IMG + j) * 2;
            out[base + 0] = oR;
            out[base + 1] = oI;
        }
    }
}

// ---------------------------------------------------------------------------
extern "C" void kernel_launch(void* const* d_in, const int* in_sizes, int n_in,
                              void* d_out, int out_size, void* d_ws, size_t ws_size,
                              hipStream_t stream) {
    const float* xR      = (const float*)d_in[0];   // (B,1,IMG,IMG)
    const float* xI      = (const float*)d_in[1];   // (B,1,IMG,IMG)
    const float* samples = (const float*)d_in[2];   // (M,2)
    const float* density = (const float*)d_in[3];   // (M,)
    float* out = (float*)d_out;                     // (B,1,IMG,IMG,2)

    // Workspace layout (floats): 4 phase planes (M*IMG each) + kw (2 * B*M)
    float* ws  = (float*)d_ws;
    float* AxR = ws;
    float* AxI = AxR + (size_t)M * IMG;
    float* AyR = AxI + (size_t)M * IMG;
    float* AyI = AyR + (size_t)M * IMG;
    float* kwR = AyI + (size_t)M * IMG;
    float* kwI = kwR + (size_t)B * M;

    // 1) phase matrices
    phase_kernel<<<(M * IMG + 255) / 256, 256, 0, stream>>>(samples, AxR, AxI, AyR, AyI);

    // 2) forward NDFT + density -> kw   (7680 waves, 8 waves/block)
    int fwdWaves = B * MT;
    fwd_kernel<<<(fwdWaves * 32) / 256, 256, 0, stream>>>(
        xR, xI, AxR, AxI, AyR, AyI, density, kwR, kwI);

    // 3) adjoint NDFT -> out            (8192 waves, 8 waves/block)
    int adjWaves = B * 16 * 8;
    adj_kernel<<<(adjWaves * 32) / 256, 256, 0, stream>>>(
        AxR, AxI, AyR, AyI, kwR, kwI, out);
}